// WaveSpectrum_65463891526209
// MI455X (gfx1250) — compile-verified
//
#include <hip/hip_runtime.h>

#define NBV     2.16e-2f
#define DZ      250.0f
#define NMODES  20
#define KS_WAVE 3.14159265358979e-7f   // 2*2*pi/4e7

typedef float v2f __attribute__((ext_vector_type(2)));
typedef float v8f __attribute__((ext_vector_type(8)));

// g_func: NBV * alpha / (ks * (c - u)^2)
__device__ __forceinline__ float g_func(float alpha, float cu) {
    return NBV * alpha / (KS_WAVE * cu * cu);
}

// ---------------------------------------------------------------------------
// Kernel 1: per-mode cumulative trapezoid + amps*exp(-cum).
// One 1024-thread block per mode; each thread owns 16 contiguous levels.
// partial[m][i] = amps[m] * exp(-cum[m][i])
// ---------------------------------------------------------------------------
__global__ void __launch_bounds__(1024)
mode_scan_kernel(const float* __restrict__ u, const float* __restrict__ alpha,
                 const float* __restrict__ amps, float* __restrict__ partial, int n) {
    const int m   = blockIdx.x;
    const int tid = threadIdx.x;
    const int i0  = tid * 16;                  // n == 16384, 1024 threads * 16

    // cs = [-100,-90,...,-10, 10,20,...,100]
    const float c = (m < 10) ? (-100.0f + 10.0f * (float)m)
                             : (10.0f * (float)(m - 9));

    float pref[16];
    float gprev = (i0 > 0) ? g_func(alpha[i0 - 1], c - u[i0 - 1]) : 0.0f;
    float run = 0.0f;
#pragma unroll
    for (int j = 0; j < 16; ++j) {
        const int i = i0 + j;
        const float gi  = g_func(alpha[i], c - u[i]);
        const float inc = (i > 0) ? (0.5f * DZ) * (gi + gprev) : 0.0f;
        run += inc;
        pref[j] = run;
        gprev = gi;
    }

    // Block-wide Hillis-Steele inclusive scan of per-thread totals in LDS.
    __shared__ float s[1024];
    s[tid] = run;
    __syncthreads();
    for (int off = 1; off < 1024; off <<= 1) {
        const float v = (tid >= off) ? s[tid - off] : 0.0f;
        __syncthreads();
        s[tid] += v;
        __syncthreads();
    }
    const float excl = s[tid] - run;           // exclusive prefix for this chunk

    const float A = amps[m];
#pragma unroll
    for (int j = 0; j < 16; ++j) {
        partial[(size_t)m * n + i0 + j] = A * expf(-(excl + pref[j]));
    }
}

// ---------------------------------------------------------------------------
// Kernel 2: Ftot[i] = sum over 20 modes of partial[m][i]
// ---------------------------------------------------------------------------
__global__ void sum_modes_kernel(const float* __restrict__ partial,
                                 float* __restrict__ F, int n) {
    const int i = blockIdx.x * blockDim.x + threadIdx.x;
    if (i >= n) return;
    float acc = 0.0f;
#pragma unroll
    for (int m = 0; m < NMODES; ++m) acc += partial[(size_t)m * n + i];
    F[i] = acc;
}

// ---------------------------------------------------------------------------
// Kernel 3: out = (D1 @ F) * rho[0] / rho   via V_WMMA_F32_16X16X4_F32.
// Block = 256 threads (8 waves) owns 16 rows; waves split K.
// A tile 16x4 (ISA layout): lanes 0-15 -> K=k,k+1 ; lanes 16-31 -> K=k+2,k+3.
// B tile 4x16 with every column = F[k..k+3] -> all 16 columns of C equal y.
// ---------------------------------------------------------------------------
__global__ void __launch_bounds__(256)
gemv_wmma_kernel(const float* __restrict__ D1, const float* __restrict__ F,
                 const float* __restrict__ rho, float* __restrict__ out, int n) {
    const int tid  = threadIdx.x;
    const int wave = tid >> 5;
    const int lane = tid & 31;
    const int half = lane >> 4;        // 0: K pair {0,1}, 1: K pair {2,3}
    const int r    = lane & 15;        // row within 16-row tile (A layout M=lane%16)
    const int rowBase = blockIdx.x * 16;

    const int kspan = n >> 3;          // K range per wave (8 waves)
    const int k0    = wave * kspan;

    const float* __restrict__ rowPtr = D1 + (size_t)(rowBase + r) * n;

    v8f acc = {};
    for (int k = k0; k < k0 + kspan; k += 16) {
        // prefetch the stream ~4KB ahead (emits global_prefetch_b8)
        __builtin_prefetch(rowPtr + k + 1024, 0, 1);
#pragma unroll
        for (int kk = 0; kk < 16; kk += 4) {
            const int kc = k + kk + 2 * half;              // even -> 8B aligned
            v2f a = *(const v2f*)(rowPtr + kc);            // A: D1[row][kc..kc+1]
            v2f b = *(const v2f*)(F + kc);                 // B: broadcast F values
            acc = __builtin_amdgcn_wmma_f32_16x16x4_f32(
                false, a, false, b, (short)0, acc, false, false);
        }
    }

    // C layout: VGPR i holds M=i (lanes 0-15) / M=i+8 (lanes 16-31); all N equal.
    // Lane 0 carries y[0..7], lane 16 carries y[8..15].
    __shared__ float red[8][16];
    if (r == 0) {
        const int mbase = half * 8;
#pragma unroll
        for (int i = 0; i < 8; ++i) red[wave][mbase + i] = acc[i];
    }
    __syncthreads();

    if (tid < 16) {
        float ysum = 0.0f;
#pragma unroll
        for (int w = 0; w < 8; ++w) ysum += red[w][tid];
        const float r0 = rho[0];
        out[rowBase + tid] = ysum * r0 / rho[rowBase + tid];
    }
}

// ---------------------------------------------------------------------------
extern "C" void kernel_launch(void* const* d_in, const int* in_sizes, int n_in,
                              void* d_out, int out_size, void* d_ws, size_t ws_size,
                              hipStream_t stream) {
    const float* u     = (const float*)d_in[0];
    const float* rho   = (const float*)d_in[1];
    const float* alpha = (const float*)d_in[2];
    const float* D1    = (const float*)d_in[3];
    const float* amps  = (const float*)d_in[4];
    float* out = (float*)d_out;
    const int n = in_sizes[0];                 // 16384

    // workspace layout: [NMODES*n] per-mode partials, then [n] Ftot
    float* partial = (float*)d_ws;
    float* F       = partial + (size_t)NMODES * n;

    mode_scan_kernel<<<NMODES, 1024, 0, stream>>>(u, alpha, amps, partial, n);
    sum_modes_kernel<<<(n + 255) / 256, 256, 0, stream>>>(partial, F, n);
    gemv_wmma_kernel<<<n / 16, 256, 0, stream>>>(D1, F, rho, out, n);
}